// GNNComplete_1778116460575
// MI455X (gfx1250) — compile-verified
//
#include <hip/hip_runtime.h>

#define N_NODES 50000
#define N_EDGES 800000
#define DIM     256
#define DIM2    512
#define L_LAYERS 5
#define BN_EPS_F 1e-5f

typedef float v2f __attribute__((ext_vector_type(2)));
typedef float v8f __attribute__((ext_vector_type(8)));

__device__ __forceinline__ float4 f4add(float4 a, float4 b) {
    return make_float4(a.x + b.x, a.y + b.y, a.z + b.z, a.w + b.w);
}

// ---------------- AtomEncoder: h[n,d] = sum_f atom_emb[f, x[n,f], d] ----------------
// 4 nodes / block, float4 per thread (2x global_load_b128-class traffic per row slice)
__global__ __launch_bounds__(256) void atom_encode_kernel(
    const int* __restrict__ x, const float* __restrict__ atom_emb,
    float* __restrict__ h)
{
    int n  = blockIdx.x * 4 + (threadIdx.x >> 6);
    int d4 = threadIdx.x & 63;                       // float4 index within the 256-wide row
    const float4* emb = (const float4*)atom_emb;
    float4 s = make_float4(0.f, 0.f, 0.f, 0.f);
#pragma unroll
    for (int f = 0; f < 9; ++f) {
        int idx = x[n * 9 + f];
        s = f4add(s, emb[((size_t)f * 120 + idx) * 64 + d4]);
    }
    ((float4*)h)[(size_t)n * 64 + d4] = s;
}

// ---------------- z = (1 + eps[l]) * h  (float4) ----------------
__global__ __launch_bounds__(256) void init_z_kernel(
    const float* __restrict__ h, const float* __restrict__ eps, int layer,
    float* __restrict__ z)
{
    size_t i = (size_t)blockIdx.x * 256 + threadIdx.x;   // float4 units
    float s = 1.0f + eps[layer];
    float4 v = ((const float4*)h)[i];
    ((float4*)z)[i] = make_float4(s * v.x, s * v.y, s * v.z, s * v.w);
}

// ---------------- edge phase: one wave32 per edge ----------------
// Each lane owns 8 contiguous floats -> 2x b128 loads per source array,
// msg = relu(h[src] + bond sums); 8 scalar f32 atomics into z[dst] (L2-resident).
__global__ __launch_bounds__(256) void edge_msg_kernel(
    const int* __restrict__ edge_index, const int* __restrict__ edge_attr,
    const float* __restrict__ bond_emb_l, const float* __restrict__ h,
    float* __restrict__ z)
{
    int wv   = (blockIdx.x * 256 + threadIdx.x) >> 5;    // edge id (grid covers E exactly)
    int lane = threadIdx.x & 31;
    int src = edge_index[wv];
    int dst = edge_index[N_EDGES + wv];
    const float4* b0 = (const float4*)(bond_emb_l + (size_t)(0 * 6 + edge_attr[wv * 3 + 0]) * DIM);
    const float4* b1 = (const float4*)(bond_emb_l + (size_t)(1 * 6 + edge_attr[wv * 3 + 1]) * DIM);
    const float4* b2 = (const float4*)(bond_emb_l + (size_t)(2 * 6 + edge_attr[wv * 3 + 2]) * DIM);
    const float4* hs = (const float4*)(h + (size_t)src * DIM);
    float*        zd = z + (size_t)dst * DIM + lane * 8;
    int q0 = lane * 2, q1 = lane * 2 + 1;

    float4 m0 = f4add(f4add(hs[q0], b0[q0]), f4add(b1[q0], b2[q0]));
    float4 m1 = f4add(f4add(hs[q1], b0[q1]), f4add(b1[q1], b2[q1]));

    atomicAdd(zd + 0, fmaxf(m0.x, 0.f));
    atomicAdd(zd + 1, fmaxf(m0.y, 0.f));
    atomicAdd(zd + 2, fmaxf(m0.z, 0.f));
    atomicAdd(zd + 3, fmaxf(m0.w, 0.f));
    atomicAdd(zd + 4, fmaxf(m1.x, 0.f));
    atomicAdd(zd + 5, fmaxf(m1.y, 0.f));
    atomicAdd(zd + 6, fmaxf(m1.z, 0.f));
    atomicAdd(zd + 7, fmaxf(m1.w, 0.f));
}

// ---------------- GEMM1: H1[N,512] = Z[N,256] @ W[256,512] + bias; fused BN stats ----
// 4 waves / block; each wave: one 16x16 tile via V_WMMA_F32_16X16X4_F32, K-step 4.
__global__ __launch_bounds__(128) void gemm1_wmma_kernel(
    const float* __restrict__ Z, const float* __restrict__ W,
    const float* __restrict__ bias, float* __restrict__ H1,
    float* __restrict__ sum, float* __restrict__ sumsq)
{
    int row0 = blockIdx.x * 16;
    int wave = threadIdx.x >> 5, lane = threadIdx.x & 31;
    int col0 = blockIdx.y * 64 + wave * 16;
    int half = lane >> 4, l16 = lane & 15;
    const float* arow = Z + (size_t)(row0 + l16) * DIM;   // A row for this lane
    int bcol = col0 + l16;                                // B/C column for this lane
    v8f acc = {};
    for (int k0 = 0; k0 < DIM; k0 += 4) {
        int ka = k0 + 2 * half;                 // lanes 0-15: K={k0,k0+1}; 16-31: K={k0+2,k0+3}
        v2f a; a.x = arow[ka];                    a.y = arow[ka + 1];
        v2f b; b.x = W[(size_t)ka * DIM2 + bcol]; b.y = W[(size_t)(ka + 1) * DIM2 + bcol];
        acc = __builtin_amdgcn_wmma_f32_16x16x4_f32(
            false, a, false, b, (short)0, acc, false, false);
    }
    float bv = bias[bcol];
    float ps = 0.f, pq = 0.f;
#pragma unroll
    for (int i = 0; i < 8; ++i) {               // C VGPR i: rows i (lanes 0-15) / i+8 (16-31)
        int r = row0 + i + 8 * half;
        float v = acc[i] + bv;
        H1[(size_t)r * DIM2 + bcol] = v;
        ps += v; pq += v * v;
    }
    atomicAdd(&sum[bcol], ps);
    atomicAdd(&sumsq[bcol], pq);
}

// ---- BN finalize: a = g*rsqrt(var+eps), c = beta - mean*a  (tiny kernel) ----
__global__ void bn_finalize_kernel(
    const float* __restrict__ sum, const float* __restrict__ sumsq,
    const float* __restrict__ gamma, const float* __restrict__ beta,
    float* __restrict__ a, float* __restrict__ c, int width)
{
    int i = blockIdx.x * 256 + threadIdx.x;
    if (i >= width) return;
    const float invN = 1.0f / (float)N_NODES;
    float m   = sum[i] * invN;
    float var = sumsq[i] * invN - m * m;
    float av  = gamma[i] * __frsqrt_rn(var + BN_EPS_F);
    a[i] = av;
    c[i] = beta[i] - m * av;
}

// ---------------- GEMM2: H2[N,256] = relu(BN1(H1)) @ W[512,256] + bias; fused BN2 stats
__global__ __launch_bounds__(128) void gemm2_wmma_kernel(
    const float* __restrict__ H1, const float* __restrict__ W,
    const float* __restrict__ bias, const float* __restrict__ a1,
    const float* __restrict__ c1, float* __restrict__ H2,
    float* __restrict__ sum, float* __restrict__ sumsq)
{
    int row0 = blockIdx.x * 16;
    int wave = threadIdx.x >> 5, lane = threadIdx.x & 31;
    int col0 = blockIdx.y * 64 + wave * 16;
    int half = lane >> 4, l16 = lane & 15;
    const float* arow = H1 + (size_t)(row0 + l16) * DIM2;
    int bcol = col0 + l16;
    v8f acc = {};
    for (int k0 = 0; k0 < DIM2; k0 += 4) {
        int ka = k0 + 2 * half;
        float x0 = fmaxf(arow[ka]     * a1[ka]     + c1[ka],     0.f);  // BN1 + ReLU on the fly
        float x1 = fmaxf(arow[ka + 1] * a1[ka + 1] + c1[ka + 1], 0.f);
        v2f a; a.x = x0; a.y = x1;
        v2f b; b.x = W[(size_t)ka * DIM + bcol]; b.y = W[(size_t)(ka + 1) * DIM + bcol];
        acc = __builtin_amdgcn_wmma_f32_16x16x4_f32(
            false, a, false, b, (short)0, acc, false, false);
    }
    float bv = bias[bcol];
    float ps = 0.f, pq = 0.f;
#pragma unroll
    for (int i = 0; i < 8; ++i) {
        int r = row0 + i + 8 * half;
        float v = acc[i] + bv;
        H2[(size_t)r * DIM + bcol] = v;
        ps += v; pq += v * v;
    }
    atomicAdd(&sum[bcol], ps);
    atomicAdd(&sumsq[bcol], pq);
}

// ---------------- out = BN2(z) (+ReLU unless last layer), float4 ----------------
__global__ __launch_bounds__(256) void apply_bn_kernel(
    const float* __restrict__ in, const float* __restrict__ a,
    const float* __restrict__ c, float* __restrict__ out, int do_relu)
{
    size_t i = (size_t)blockIdx.x * 256 + threadIdx.x;   // float4 units
    int cb = threadIdx.x & 63;                           // column float4 index (256/4)
    float4 v  = ((const float4*)in)[i];
    float4 aa = ((const float4*)a)[cb];
    float4 cc = ((const float4*)c)[cb];
    v = make_float4(v.x * aa.x + cc.x, v.y * aa.y + cc.y,
                    v.z * aa.z + cc.z, v.w * aa.w + cc.w);
    if (do_relu)
        v = make_float4(fmaxf(v.x, 0.f), fmaxf(v.y, 0.f),
                        fmaxf(v.z, 0.f), fmaxf(v.w, 0.f));
    ((float4*)out)[i] = v;
}

extern "C" void kernel_launch(void* const* d_in, const int* in_sizes, int n_in,
                              void* d_out, int out_size, void* d_ws, size_t ws_size,
                              hipStream_t stream)
{
    const int*   x          = (const int*)d_in[0];
    const int*   edge_index = (const int*)d_in[1];
    const int*   edge_attr  = (const int*)d_in[2];
    const float* atom_emb   = (const float*)d_in[3];
    const float* bond_emb   = (const float*)d_in[4];
    const float* eps        = (const float*)d_in[5];
    const float* W1  = (const float*)d_in[6];
    const float* b1  = (const float*)d_in[7];
    const float* g1  = (const float*)d_in[8];
    const float* be1 = (const float*)d_in[9];
    const float* W2  = (const float*)d_in[10];
    const float* b2  = (const float*)d_in[11];
    const float* g2  = (const float*)d_in[12];
    const float* be2 = (const float*)d_in[13];
    float* out = (float*)d_out;

    float* ws = (float*)d_ws;
    float* h  = ws;                               // N*256
    float* z  = h  + (size_t)N_NODES * DIM;       // N*256 (agg, then reused as H2)
    float* h1 = z  + (size_t)N_NODES * DIM;       // N*512
    float* st = h1 + (size_t)N_NODES * DIM2;      // 3072 floats of stats
    float* sum1 = st,        *sq1 = st + 512,  *a1 = st + 1024, *c1 = st + 1536;
    float* sum2 = st + 2048, *sq2 = st + 2304, *a2 = st + 2560, *c2 = st + 2816;

    const int ELEM4_BLOCKS = (N_NODES * 64) / 256;   // N*DIM float4s / 256 threads = 12500

    atom_encode_kernel<<<N_NODES / 4, 256, 0, stream>>>(x, atom_emb, h);

    for (int l = 0; l < L_LAYERS; ++l) {
        init_z_kernel<<<ELEM4_BLOCKS, 256, 0, stream>>>(h, eps, l, z);
        edge_msg_kernel<<<N_EDGES / 8, 256, 0, stream>>>(
            edge_index, edge_attr, bond_emb + (size_t)l * 3 * 6 * DIM, h, z);

        hipMemsetAsync(st, 0, 3072 * sizeof(float), stream);

        gemm1_wmma_kernel<<<dim3(N_NODES / 16, DIM2 / 64), 128, 0, stream>>>(
            z, W1 + (size_t)l * DIM * DIM2, b1 + (size_t)l * DIM2, h1, sum1, sq1);
        bn_finalize_kernel<<<2, 256, 0, stream>>>(
            sum1, sq1, g1 + (size_t)l * DIM2, be1 + (size_t)l * DIM2, a1, c1, DIM2);

        gemm2_wmma_kernel<<<dim3(N_NODES / 16, DIM / 64), 128, 0, stream>>>(
            h1, W2 + (size_t)l * DIM2 * DIM, b2 + (size_t)l * DIM, a1, c1, z, sum2, sq2);
        bn_finalize_kernel<<<1, 256, 0, stream>>>(
            sum2, sq2, g2 + (size_t)l * DIM, be2 + (size_t)l * DIM, a2, c2, DIM);

        float* dst = (l == L_LAYERS - 1) ? out : h;
        apply_bn_kernel<<<ELEM4_BLOCKS, 256, 0, stream>>>(
            z, a2, c2, dst, (l < L_LAYERS - 1) ? 1 : 0);
    }
}